// CorePatternLearner_6365141532704
// MI455X (gfx1250) — compile-verified
//
#include <hip/hip_runtime.h>
#include <hip/hip_bf16.h>
#include <math.h>

typedef __attribute__((ext_vector_type(2))) float v2f;
typedef __attribute__((ext_vector_type(8))) float v8f;

#define NEG_SLOPE 0.2f

// ---------------- init kernels ----------------
__global__ void zero_f32(float* __restrict__ p, long long n) {
  long long i = (long long)blockIdx.x * blockDim.x + threadIdx.x;
  if (i < n) p[i] = 0.0f;
}

__global__ void fill_i32(int* __restrict__ p, long long n, int v) {
  long long i = (long long)blockIdx.x * blockDim.x + threadIdx.x;
  if (i < n) p[i] = v;
}

// ---------------- dense GEMM via V_WMMA_F32_16X16X4_F32 ----------------
// Hout[M, TN*16] = X[M, 128] @ W[128, TN*16].   M % 16 == 0.
// Block = 256 threads = 8 waves.  The block stages all of W into LDS
// *transposed* (Wl[n*128 + k]) so each B fragment is one contiguous
// ds_load_b64.  Each wave owns one 16-row panel and accumulates all TN
// n-tiles (TN*8 acc VGPRs), so one A load (global b64) feeds TN WMMAs.
// EXEC is all-1s in every active wave (guards are wave-uniform), as WMMA
// requires.
template <int TN>
__global__ __launch_bounds__(256) void gemm_wmma_f32(const float* __restrict__ X,
                                                     const float* __restrict__ W,
                                                     float* __restrict__ Hout,
                                                     int M) {
  constexpr int KC = 128;
  constexpr int NC = TN * 16;
  __shared__ float Wl[KC * NC];  // transposed: Wl[n*KC + k]

  // ---- stage W (row-major global reads, transposed LDS writes) ----
  for (int idx = threadIdx.x; idx < KC * NC; idx += 256) {
    int k = idx / NC;
    int n = idx - k * NC;
    Wl[n * KC + k] = W[idx];
  }
  __syncthreads();

  const int wave = threadIdx.x >> 5;
  const int lane = threadIdx.x & 31;
  const int mt = blockIdx.x * 8 + wave;
  if (mt * 16 >= M) return;              // wave-uniform

  const int half = lane >> 4;            // 0: K pair {0,1}, 1: K pair {2,3}
  const int l15  = lane & 15;
  const float* Xrow = X + (long long)(mt * 16 + l15) * KC;

  v8f acc[TN];
#pragma unroll
  for (int t = 0; t < TN; ++t) acc[t] = (v8f){};

  for (int k = 0; k < KC; k += 4) {
    // A 16x4 tile: lane (l15,half) holds X[row][k+2*half .. +1]
    v2f a = *(const v2f*)(Xrow + k + 2 * half);
#pragma unroll
    for (int t = 0; t < TN; ++t) {
      // B 4x16 tile: lane holds W[k+2*half + {0,1}][t*16+l15] (contig in Wl)
      v2f b = *(const v2f*)(Wl + (t * 16 + l15) * KC + k + 2 * half);
      acc[t] = __builtin_amdgcn_wmma_f32_16x16x4_f32(
          /*neg_a=*/false, a, /*neg_b=*/false, b,
          /*c_mod=*/(short)0, acc[t], /*reuse_a=*/false, /*reuse_b=*/false);
    }
  }

  // D 16x16: VGPR r -> row (r + 8*half), col l15
#pragma unroll
  for (int t = 0; t < TN; ++t) {
#pragma unroll
    for (int r = 0; r < 8; ++r) {
      Hout[(long long)(mt * 16 + r + 8 * half) * NC + t * 16 + l15] = acc[t][r];
    }
  }
}

// ---------------- per-node attention scores ----------------
// es[n,h] = sum_c Hm[n,h*C+c]*a_src[h*C+c], ed likewise.
__global__ void node_scores(const float* __restrict__ Hm,
                            const float* __restrict__ a_src,
                            const float* __restrict__ a_dst,
                            float* __restrict__ es, float* __restrict__ ed,
                            int Nn, int Hh, int C) {
  long long i = (long long)blockIdx.x * blockDim.x + threadIdx.x;
  if (i >= (long long)Nn * Hh) return;
  int n = (int)(i / Hh);
  int h = (int)(i - (long long)n * Hh);
  const float* row = Hm + (long long)n * Hh * C + (long long)h * C;
  const float* as  = a_src + (long long)h * C;
  const float* ad  = a_dst + (long long)h * C;
  float s1 = 0.0f, s2 = 0.0f;
  for (int c = 0; c < C; ++c) {
    float v = row[c];
    s1 += v * as[c];
    s2 += v * ad[c];
  }
  es[i] = s1;
  ed[i] = s2;
}

// monotone float<->int encoding so signed atomicMax orders floats correctly
__device__ __forceinline__ int enc_f32(float f) {
  int b = __float_as_int(f);
  return (b >= 0) ? b : (b ^ 0x7FFFFFFF);
}
__device__ __forceinline__ float dec_f32(int e) {
  int b = (e >= 0) ? e : (e ^ 0x7FFFFFFF);
  return __int_as_float(b);
}

// ---------------- edge pass 1: logits + segment max ----------------
// edge ids e in [0,E) are real edges, e in [E, E+Nn) are self-loops.
__global__ void edge_logit_max(const long long* __restrict__ ei, int E, int Nn, int Hh,
                               const float* __restrict__ es, const float* __restrict__ ed,
                               float* __restrict__ ebuf, int* __restrict__ mx) {
  long long t = (long long)blockIdx.x * blockDim.x + threadIdx.x;
  long long total = (long long)(E + Nn) * Hh;
  if (t >= total) return;
  int e = (int)(t / Hh);
  int h = (int)(t - (long long)e * Hh);
  int s, d;
  if (e < E) { s = (int)ei[e]; d = (int)ei[(long long)E + e]; }
  else       { s = e - E; d = s; }
  float z = es[(long long)s * Hh + h] + ed[(long long)d * Hh + h];
  z = (z > 0.0f) ? z : NEG_SLOPE * z;          // leaky_relu
  ebuf[t] = z;
  atomicMax(mx + (long long)d * Hh + h, enc_f32(z));
}

// ---------------- edge pass 2: exp(z - mx) + segment sum ----------------
__global__ void edge_exp_sum(const long long* __restrict__ ei, int E, int Nn, int Hh,
                             float* __restrict__ ebuf, const int* __restrict__ mx,
                             float* __restrict__ den) {
  long long t = (long long)blockIdx.x * blockDim.x + threadIdx.x;
  long long total = (long long)(E + Nn) * Hh;
  if (t >= total) return;
  int e = (int)(t / Hh);
  int h = (int)(t - (long long)e * Hh);
  int d = (e < E) ? (int)ei[(long long)E + e] : (e - E);
  float m  = dec_f32(mx[(long long)d * Hh + h]);
  float ex = expf(ebuf[t] - m);
  ebuf[t] = ex;
  atomicAdd(den + (long long)d * Hh + h, ex);
}

// ---------------- edge pass 3a: normalize -> alpha (in place) ----------------
__global__ void edge_alpha(const long long* __restrict__ ei, int E, int Nn, int Hh,
                           float* __restrict__ ebuf, const float* __restrict__ den) {
  long long t = (long long)blockIdx.x * blockDim.x + threadIdx.x;
  long long total = (long long)(E + Nn) * Hh;
  if (t >= total) return;
  int e = (int)(t / Hh);
  int h = (int)(t - (long long)e * Hh);
  int d = (e < E) ? (int)ei[(long long)E + e] : (e - E);
  ebuf[t] = ebuf[t] / (den[(long long)d * Hh + h] + 1e-16f);
}

// ---------------- edge pass 3b: weighted scatter ----------------
// One thread per (edge, feature j), j in [0, Hh*C); head = j / C.
// Consecutive threads cover one source row -> coalesced loads, and target
// consecutive addresses of one dst row -> grouped atomics.
__global__ void edge_scatter(const long long* __restrict__ ei, int E, int Nn,
                             int Hh, int C,
                             const float* __restrict__ alpha,
                             const float* __restrict__ Hfeat,
                             float* __restrict__ agg) {
  const int F = Hh * C;
  long long t = (long long)blockIdx.x * blockDim.x + threadIdx.x;
  long long total = (long long)(E + Nn) * F;
  if (t >= total) return;
  int e = (int)(t / F);
  int j = (int)(t - (long long)e * F);
  int s, d;
  if (e < E) { s = (int)ei[e]; d = (int)ei[(long long)E + e]; }
  else       { s = e - E; d = s; }
  int h = j / C;
  float a = alpha[(long long)e * Hh + h];
  atomicAdd(agg + (long long)d * F + j, a * Hfeat[(long long)s * F + j]);
}

// ---------------- epilogues ----------------
__global__ void bias_elu_inplace(float* __restrict__ agg, const float* __restrict__ bias,
                                 int Nn, int F) {
  long long i = (long long)blockIdx.x * blockDim.x + threadIdx.x;
  if (i >= (long long)Nn * F) return;
  int j = (int)(i % F);
  float v = agg[i] + bias[j];
  agg[i] = (v > 0.0f) ? v : (expf(v) - 1.0f);   // elu
}

__global__ void bias_add_inplace(float* __restrict__ out, const float* __restrict__ bias,
                                 int Nn, int F) {
  long long i = (long long)blockIdx.x * blockDim.x + threadIdx.x;
  if (i >= (long long)Nn * F) return;
  int j = (int)(i % F);
  out[i] += bias[j];
}

// ---------------- host launch ----------------
static inline unsigned cdiv_ll(long long a, int b) {
  return (unsigned)((a + b - 1) / b);
}

extern "C" void kernel_launch(void* const* d_in, const int* in_sizes, int n_in,
                              void* d_out, int out_size, void* d_ws, size_t ws_size,
                              hipStream_t stream) {
  const float*     x   = (const float*)d_in[0];
  const long long* ei  = (const long long*)d_in[1];   // int64 edge_index [2,E]
  const float*     W1  = (const float*)d_in[2];
  const float*     as1 = (const float*)d_in[3];
  const float*     ad1 = (const float*)d_in[4];
  const float*     b1  = (const float*)d_in[5];
  const float*     W2  = (const float*)d_in[6];
  const float*     as2 = (const float*)d_in[7];
  const float*     ad2 = (const float*)d_in[8];
  const float*     b2  = (const float*)d_in[9];
  float* out = (float*)d_out;

  const int IN = 128, HEADS = 4, HID = 32, OUT = 64;
  const int F1 = HEADS * HID;                 // 128
  const int Nn = in_sizes[0] / IN;            // 100000 (divisible by 16)
  const int E  = in_sizes[1] / 2;             // 1600000
  const long long Etot = (long long)E + Nn;   // with self-loops

  // ---- workspace carve-out (floats) ----
  float* ws = (float*)d_ws;
  size_t o = 0;
  float* h1   = ws + o; o += (size_t)Nn * F1;   // layer1 features; reused as h2
  float* agg1 = ws + o; o += (size_t)Nn * F1;   // layer1 agg; in-place elu -> layer2 input
  float* es1  = ws + o; o += (size_t)Nn * HEADS;
  float* ed1  = ws + o; o += (size_t)Nn * HEADS;
  int*   mx1  = (int*)(ws + o); o += (size_t)Nn * HEADS;
  float* den1 = ws + o; o += (size_t)Nn * HEADS;
  float* es2  = ws + o; o += (size_t)Nn;
  float* ed2  = ws + o; o += (size_t)Nn;
  int*   mx2  = (int*)(ws + o); o += (size_t)Nn;
  float* den2 = ws + o; o += (size_t)Nn;
  float* ebuf = ws + o; o += (size_t)Etot * HEADS;  // reused for layer2 (needs Etot*1)

  const int B = 256;
  const int INT_MIN_V = (int)0x80000000;

  // ---- init (deterministic every call) ----
  fill_i32<<<cdiv_ll((long long)Nn * HEADS, B), B, 0, stream>>>(mx1, (long long)Nn * HEADS, INT_MIN_V);
  zero_f32<<<cdiv_ll((long long)Nn * HEADS, B), B, 0, stream>>>(den1, (long long)Nn * HEADS);
  zero_f32<<<cdiv_ll((long long)Nn * F1, B), B, 0, stream>>>(agg1, (long long)Nn * F1);
  fill_i32<<<cdiv_ll((long long)Nn, B), B, 0, stream>>>(mx2, (long long)Nn, INT_MIN_V);
  zero_f32<<<cdiv_ll((long long)Nn, B), B, 0, stream>>>(den2, (long long)Nn);
  zero_f32<<<cdiv_ll((long long)Nn * OUT, B), B, 0, stream>>>(out, (long long)Nn * OUT);

  // ================= layer 1 =================
  {
    int tilesM = Nn / 16;                       // 6250
    gemm_wmma_f32<8><<<cdiv_ll(tilesM, 8), B, 0, stream>>>(x, W1, h1, Nn);
  }
  node_scores<<<cdiv_ll((long long)Nn * HEADS, B), B, 0, stream>>>(h1, as1, ad1, es1, ed1, Nn, HEADS, HID);
  edge_logit_max<<<cdiv_ll(Etot * HEADS, B), B, 0, stream>>>(ei, E, Nn, HEADS, es1, ed1, ebuf, mx1);
  edge_exp_sum<<<cdiv_ll(Etot * HEADS, B), B, 0, stream>>>(ei, E, Nn, HEADS, ebuf, mx1, den1);
  edge_alpha<<<cdiv_ll(Etot * HEADS, B), B, 0, stream>>>(ei, E, Nn, HEADS, ebuf, den1);
  edge_scatter<<<cdiv_ll(Etot * F1, B), B, 0, stream>>>(ei, E, Nn, HEADS, HID, ebuf, h1, agg1);
  bias_elu_inplace<<<cdiv_ll((long long)Nn * F1, B), B, 0, stream>>>(agg1, b1, Nn, F1);

  // ================= layer 2 (heads=1, concat=False -> mean over 1 head = identity) =================
  float* h2 = h1;  // reuse
  {
    int tilesM = Nn / 16;
    gemm_wmma_f32<4><<<cdiv_ll(tilesM, 8), B, 0, stream>>>(agg1, W2, h2, Nn);
  }
  node_scores<<<cdiv_ll((long long)Nn, B), B, 0, stream>>>(h2, as2, ad2, es2, ed2, Nn, 1, OUT);
  edge_logit_max<<<cdiv_ll(Etot, B), B, 0, stream>>>(ei, E, Nn, 1, es2, ed2, ebuf, mx2);
  edge_exp_sum<<<cdiv_ll(Etot, B), B, 0, stream>>>(ei, E, Nn, 1, ebuf, mx2, den2);
  edge_alpha<<<cdiv_ll(Etot, B), B, 0, stream>>>(ei, E, Nn, 1, ebuf, den2);
  edge_scatter<<<cdiv_ll(Etot * OUT, B), B, 0, stream>>>(ei, E, Nn, 1, OUT, ebuf, h2, out);
  bias_add_inplace<<<cdiv_ll((long long)Nn * OUT, B), B, 0, stream>>>(out, b2, Nn, OUT);
}